// HaarWaveletTransform3D_4715874091393
// MI455X (gfx1250) — compile-verified
//
#include <hip/hip_runtime.h>
#include <hip/hip_bf16.h>

typedef __attribute__((ext_vector_type(2))) float v2f;
typedef __attribute__((ext_vector_type(4))) float v4f;
typedef __attribute__((ext_vector_type(8))) float v8f;

// Haar bank is a scaled Hadamard: B[e,s] = 0.3536 * (-1)^popcount(e & s)
// e = i*4 + j*2 + k  (bit2 = t, bit1 = h, bit0 = w); columns s >= 8 are zero pad.
__device__ __forceinline__ float haarB(int e, int s) {
    if (s >= 8) return 0.0f;
    return (__builtin_popcount(e & s) & 1) ? -0.3536f : 0.3536f;
}

// x:   [2, 3, 33, 512, 512] fp32
// out: [2, 24, 17, 256, 256] fp32  (subband-major channels: och = s*3 + c)
//
// One wave owns (b, c, Tb, Hb) = one 256-wide output row = 16 tiles of 16
// W-blocks. Tiles are processed in groups of 4: 8 loads issued back-to-back
// (2 KB of reads in flight per wave for latency hiding), then 4 chained
// V_WMMA_F32_16X16X4_F32 pairs (K=8 = 2 x K=4), then one EXEC-masked block
// of 8 b128 stores.
__global__ __launch_bounds__(256, 1) void haar3d_wmma_kernel(
        const float* __restrict__ x, float* __restrict__ out) {
    const int lane = threadIdx.x & 31;
    const int wave = threadIdx.x >> 5;
    const int m    = lane & 15;   // A: row M. B/C/D: column N (= subband).
    const int half = lane >> 4;   // selects K pair (A/B) and M+8 half (C/D)

    unsigned gt = blockIdx.x * 8u + (unsigned)wave;
    const unsigned hb = gt & 255u; gt >>= 8;   // 256 H-block rows
    const unsigned tb = gt % 17u;  gt /= 17u;  // 17 T-blocks
    const unsigned ch = gt % 3u;
    const unsigned bb = gt / 3u;

    // Synthesize B matrices in-register (no memory traffic).
    // B layout mirrors A: lane%16 = N, VGPR0/1 = K=(0,1) lanes0-15 / K=(2,3) lanes16-31.
    const int e0 = half * 2;
    v2f blo, bhi;
    blo.x = haarB(e0 + 0, m);  blo.y = haarB(e0 + 1, m);   // chunk0: e = 0..3 (frame t0)
    bhi.x = haarB(e0 + 4, m);  bhi.y = haarB(e0 + 5, m);   // chunk1: e = 4..7 (frame t1)

    // Causal replicate pad: padded frames (2Tb, 2Tb+1) = original (max(2Tb-1,0), 2Tb)
    const int t1 = 2 * (int)tb;
    const int t0 = (tb == 0) ? 0 : (t1 - 1);

    // Input strides: W=1, H=512, T=512*512, C=33*T
    const size_t chanBase = (size_t)(bb * 3u + ch) * 33u;
    const size_t rowoff   = (size_t)(2u * hb + (unsigned)half) * 512u + 2u * (unsigned)m;
    const float* p0 = x + (chanBase + (unsigned)t0) * 262144u + rowoff;
    const float* p1 = x + (chanBase + (unsigned)t1) * 262144u + rowoff;

    // Output strides: W=1, H=256, T=256*256, CH=17*T, B=24*CH.
    // Valid only for lanes with m < 8; advances contiguously (+16 floats/tile).
    const bool writer = (m < 8);
    float* po = out + ((size_t)(bb * 24u + (unsigned)m * 3u + ch) * 17u + tb) * 65536u
                    + (size_t)hb * 256u + (unsigned)half * 8u;

    for (int g = 0; g < 4; ++g) {          // 4 groups of 4 tiles = 16 W-tiles
        v2f a0[4], a1[4];
#pragma unroll
        for (int j = 0; j < 4; ++j) {      // 8 loads issued before any consume
            a0[j] = __builtin_nontemporal_load((const v2f*)(p0 + 32 * j));
            a1[j] = __builtin_nontemporal_load((const v2f*)(p1 + 32 * j));
        }

        v8f acc[4];
#pragma unroll
        for (int j = 0; j < 4; ++j) {
            v8f c = {0.f, 0.f, 0.f, 0.f, 0.f, 0.f, 0.f, 0.f};
            c = __builtin_amdgcn_wmma_f32_16x16x4_f32(false, a0[j], false, blo,
                                                      (short)0, c, false, false);
            c = __builtin_amdgcn_wmma_f32_16x16x4_f32(false, a1[j], false, bhi,
                                                      (short)0, c, false, false);
            acc[j] = c;
        }

        // D: lane holds subband N = m for W-blocks M = r + 8*half (r = VGPR 0..7)
        // -> 8 contiguous W outputs per tile -> two b128 stores per tile.
        if (writer) {
#pragma unroll
            for (int j = 0; j < 4; ++j) {
                __builtin_nontemporal_store(
                    (v4f){acc[j][0], acc[j][1], acc[j][2], acc[j][3]},
                    (v4f*)(po + 16 * j));
                __builtin_nontemporal_store(
                    (v4f){acc[j][4], acc[j][5], acc[j][6], acc[j][7]},
                    (v4f*)(po + 16 * j + 4));
            }
        }
        p0 += 128; p1 += 128; po += 64;
    }
}

extern "C" void kernel_launch(void* const* d_in, const int* in_sizes, int n_in,
                              void* d_out, int out_size, void* d_ws, size_t ws_size,
                              hipStream_t stream) {
    const float* x  = (const float*)d_in[0];
    float* out      = (float*)d_out;
    // wave-jobs = b * c * Tb * Hout = 2*3*17*256 = 26112; 8 waves/block -> 3264 blocks
    const unsigned jobs   = 2u * 3u * 17u * 256u;
    const unsigned blocks = jobs / 8u;   // exact -> EXEC all-ones at every WMMA
    haar3d_wmma_kernel<<<blocks, 256, 0, stream>>>(x, out);
}